// LGE_Exp_32177894981907
// MI455X (gfx1250) — compile-verified
//
#include <hip/hip_runtime.h>
#include <hip/hip_bf16.h>
#include <stdint.h>

// LGE exp: out[mu] = expm(sum_i beta[mu,i] * (W_i - W_i^dag - tr/3 I)) @ U[mu]
// Sites: 64*16*16*16 = 262144, matrices 3x3 complex64 (interleaved float re,im).

#define NS   262144      // lattice sites per direction
#define TPB  128         // threads per block (4 waves, wave32)
#define SLOT 20          // floats per site slot in LDS (80 B: 16B-aligned for b128)

// ---- CDNA5 async global->LDS copy of 72 bytes (one 3x3 complex matrix) ----
// the immediate offset is added to both the LDS address (ldsOff) and the global address (base + off)
__device__ __forceinline__ void asyncCopy72(uint32_t ldsOff, const float* gbase, uint32_t gByteOff) {
    uint64_t g = (uint64_t)(uintptr_t)gbase;
    asm volatile(
        "global_load_async_to_lds_b128 %0, %1, %2 offset:0\n\t"
        "global_load_async_to_lds_b128 %0, %1, %2 offset:16\n\t"
        "global_load_async_to_lds_b128 %0, %1, %2 offset:32\n\t"
        "global_load_async_to_lds_b128 %0, %1, %2 offset:48\n\t"
        "global_load_async_to_lds_b64 %0, %1, %2 offset:64"
        :
        : "v"(ldsOff), "v"(gByteOff), "s"(g)
        : "memory");
}

#define WAIT_ASYNC(n) asm volatile("s_wait_asynccnt " #n : : : "memory")
#define WAIT_DS0()    asm volatile("s_wait_dscnt 0" : : : "memory")

// ---- load 18 floats (one 3x3 complex) from an aligned LDS slot ----
__device__ __forceinline__ void loadSite18(const float* s, float p[18]) {
    const float4* v4 = reinterpret_cast<const float4*>(s);
    float4 a = v4[0], b = v4[1], c = v4[2], d = v4[3];
    float2 e = *reinterpret_cast<const float2*>(s + 16);
    p[0]=a.x; p[1]=a.y; p[2]=a.z; p[3]=a.w;
    p[4]=b.x; p[5]=b.y; p[6]=b.z; p[7]=b.w;
    p[8]=c.x; p[9]=c.y; p[10]=c.z; p[11]=c.w;
    p[12]=d.x; p[13]=d.y; p[14]=d.z; p[15]=d.w;
    p[16]=e.x; p[17]=e.y;
}

// Q = -i * traceless(W - W^dag)  -> Hermitian traceless, stored compactly:
// q[0..2]=diag (real), q[3],q[4]=Re/Im q01, q[5],q[6]=q02, q[7],q[8]=q12
__device__ __forceinline__ void buildQ(const float* s, float q[9]) {
    float p[18];
    loadSite18(s, p);
    float s_im = p[1] + p[9] + p[17];        // Im(W00)+Im(W11)+Im(W22)
    float t = s_im * (1.0f / 3.0f);
    q[0] = 2.0f * (p[1]  - t);
    q[1] = 2.0f * (p[9]  - t);
    q[2] = 2.0f * (p[17] - t);
    q[3] = p[3]  + p[7];   q[4] = p[6]  - p[2];    // q01
    q[5] = p[5]  + p[13];  q[6] = p[12] - p[4];    // q02
    q[7] = p[11] + p[15];  q[8] = p[14] - p[10];   // q12
}

// exp(iQ) = f0 I + f1 Q + f2 Q^2   (Morningstar-Peardon analytic SU(3) exp),
// then O = f0 U + f1 (Q U) + f2 (Q (Q U)).
__device__ __forceinline__ void expmul(const float q[9], const float p[18], float o[18]) {
    const float d0 = q[0], d1 = q[1], d2 = q[2];
    const float r01 = q[3], i01 = q[4], r02 = q[5], i02 = q[6], r12 = q[7], i12 = q[8];
    const float n01 = r01*r01 + i01*i01;
    const float n02 = r02*r02 + i02*i02;
    const float n12 = r12*r12 + i12*i12;
    const float c1 = 0.5f*(d0*d0 + d1*d1 + d2*d2) + n01 + n02 + n12;
    // Re(q01 * q12 * conj(q02))
    const float xr = r01*r12 - i01*i12;
    const float xi = r01*i12 + i01*r12;
    const float tRe = xr*r02 + xi*i02;
    const float c0 = d0*d1*d2 + 2.0f*tRe - d0*n12 - d1*n02 - d2*n01;

    bool neg = (c0 < 0.0f);
    const float c0a = fabsf(c0);

    float f0r, f0i, f1r, f1i, f2r, f2i;
    if (c1 < 1e-12f) {
        // Q ~ 0: exp(iQ) = I + iQ - Q^2/2
        f0r = 1.0f; f0i = 0.0f; f1r = 0.0f; f1i = 1.0f; f2r = -0.5f; f2i = 0.0f;
        neg = false;
    } else {
        const float c1_3 = c1 * (1.0f / 3.0f);
        const float rs   = __frsqrt_rn(c1_3);              // 1/sqrt(c1/3)
        const float sc13 = c1_3 * rs;                      // sqrt(c1/3)
        // ratio = c0a / (2*(c1/3)^{3/2}) = 0.5*c0a*rs^3
        float ratio = 0.5f * c0a * rs * rs * rs;
        ratio = fminf(ratio, 1.0f);
        const float theta = acosf(ratio);                  // precise (once per site,mu)
        float sth, cth;
        __sincosf(theta * (1.0f / 3.0f), &sth, &cth);      // hw v_sin/v_cos
        const float u = sc13 * cth;
        const float w = sc13 * 1.7320508075688772f * sth;  // sqrt(c1)*sin(theta/3)
        const float u2 = u*u, w2 = w*w;
        float sw, cw;
        __sincosf(w, &sw, &cw);                            // hw v_sin/v_cos
        float xi0;
        if (w2 > 0.0025f) xi0 = sw / w;
        else xi0 = 1.0f - w2*(1.0f/6.0f)*(1.0f - w2*0.05f*(1.0f - w2*(1.0f/42.0f)));
        float su, cu;
        __sincosf(u, &su, &cu);                            // e^{-iu} = (cu,-su)
        const float s2u = 2.0f*su*cu;                      // e^{2iu} via double angle
        const float c2u = 1.0f - 2.0f*su*su;
        const float A = 8.0f*u2*cw;
        const float B = 2.0f*u*(3.0f*u2 + w2)*xi0;
        const float h0r = (u2 - w2)*c2u + cu*A + su*B;
        const float h0i = (u2 - w2)*s2u + cu*B - su*A;
        const float C = 2.0f*u*cw;
        const float D = (3.0f*u2 - w2)*xi0;
        const float h1r = 2.0f*u*c2u - (cu*C - su*D);
        const float h1i = 2.0f*u*s2u + (cu*D + su*C);
        const float E = cw;
        const float F = 3.0f*u*xi0;
        const float h2r = c2u - (cu*E + su*F);
        const float h2i = s2u - (cu*F - su*E);
        float den = 9.0f*u2 - w2;
        den += (den >= 0.0f ? 1e-35f : -1e-35f);
        const float rd = 1.0f / den;
        f0r = h0r*rd; f0i = h0i*rd;
        f1r = h1r*rd; f1i = h1i*rd;
        f2r = h2r*rd; f2i = h2i*rd;
    }
    if (neg) { f0i = -f0i; f1r = -f1r; f2i = -f2i; }  // f_j(-c0) = (-1)^j conj f_j(c0)

    // Hermitian Q in full form
    const float Qr[3][3] = {{d0, r01, r02}, {r01, d1, r12}, {r02, r12, d2}};
    const float Qi[3][3] = {{0.0f, i01, i02}, {-i01, 0.0f, i12}, {-i02, -i12, 0.0f}};
    float Ur[3][3], Ui[3][3];
#pragma unroll
    for (int a = 0; a < 3; a++)
#pragma unroll
        for (int b = 0; b < 3; b++) {
            Ur[a][b] = p[2*(3*a + b)];
            Ui[a][b] = p[2*(3*a + b) + 1];
        }
    float Vr[3][3], Vi[3][3], Wr[3][3], Wi[3][3];
#pragma unroll
    for (int a = 0; a < 3; a++)
#pragma unroll
        for (int b = 0; b < 3; b++) {
            float ar = 0.0f, ai = 0.0f;
#pragma unroll
            for (int k = 0; k < 3; k++) {
                ar += Qr[a][k]*Ur[k][b] - Qi[a][k]*Ui[k][b];
                ai += Qr[a][k]*Ui[k][b] + Qi[a][k]*Ur[k][b];
            }
            Vr[a][b] = ar; Vi[a][b] = ai;
        }
#pragma unroll
    for (int a = 0; a < 3; a++)
#pragma unroll
        for (int b = 0; b < 3; b++) {
            float ar = 0.0f, ai = 0.0f;
#pragma unroll
            for (int k = 0; k < 3; k++) {
                ar += Qr[a][k]*Vr[k][b] - Qi[a][k]*Vi[k][b];
                ai += Qr[a][k]*Vi[k][b] + Qi[a][k]*Vr[k][b];
            }
            Wr[a][b] = ar; Wi[a][b] = ai;
        }
#pragma unroll
    for (int a = 0; a < 3; a++)
#pragma unroll
        for (int b = 0; b < 3; b++) {
            o[2*(3*a + b)]     = f0r*Ur[a][b] - f0i*Ui[a][b] + f1r*Vr[a][b] - f1i*Vi[a][b]
                               + f2r*Wr[a][b] - f2i*Wi[a][b];
            o[2*(3*a + b) + 1] = f0r*Ui[a][b] + f0i*Ur[a][b] + f1r*Vi[a][b] + f1i*Vr[a][b]
                               + f2r*Wi[a][b] + f2i*Wr[a][b];
        }
}

__device__ __forceinline__ void doMu(const float Q[4][9], const float* beta16, int mu,
                                     const float* sU, float* outp) {
    const float b0 = beta16[4*mu + 0], b1 = beta16[4*mu + 1];
    const float b2 = beta16[4*mu + 2], b3 = beta16[4*mu + 3];
    float q[9];
#pragma unroll
    for (int k = 0; k < 9; k++)
        q[k] = b0*Q[0][k] + b1*Q[1][k] + b2*Q[2][k] + b3*Q[3][k];
    float pU[18], o[18];
    loadSite18(sU, pU);
    expmul(q, pU, o);
#pragma unroll
    for (int k = 0; k < 9; k++) {
        float2 v = make_float2(o[2*k], o[2*k + 1]);
        *reinterpret_cast<float2*>(outp + 2*k) = v;   // b64 stores, 8B-aligned
    }
}

__global__ __launch_bounds__(TPB)
void lge_exp_kernel(const float* __restrict__ U, const float* __restrict__ Wg,
                    const float* __restrict__ beta, float* __restrict__ out) {
    // 6 buffers: W0..W3, U ping, U pong
    __shared__ __align__(16) float smem[6 * TPB * SLOT];   // 60 KB
    const int tid  = threadIdx.x;
    const int site = blockIdx.x * TPB + tid;
    const uint32_t gOff = (uint32_t)site * 72u;            // byte offset of this site's matrix
    const uint32_t ldsBase   = (uint32_t)(uintptr_t)(&smem[0]);
    const uint32_t slotB     = ldsBase + (uint32_t)tid * (SLOT * 4);
    const uint32_t bufStride = (uint32_t)(TPB * SLOT * 4); // 10240 B per buffer

    // Stage W0..W3 and U0 through the async DMA pipe (5 async instrs each).
    asyncCopy72(slotB + 0u*bufStride, Wg + 0*(size_t)NS*18, gOff);
    asyncCopy72(slotB + 1u*bufStride, Wg + 1*(size_t)NS*18, gOff);
    asyncCopy72(slotB + 2u*bufStride, Wg + 2*(size_t)NS*18, gOff);
    asyncCopy72(slotB + 3u*bufStride, Wg + 3*(size_t)NS*18, gOff);
    asyncCopy72(slotB + 4u*bufStride, U  + 0*(size_t)NS*18, gOff);   // outstanding: 25

    float Q[4][9];
    WAIT_ASYNC(20); buildQ(&smem[0*TPB*SLOT + tid*SLOT], Q[0]);
    WAIT_ASYNC(15); buildQ(&smem[1*TPB*SLOT + tid*SLOT], Q[1]);
    WAIT_ASYNC(10); buildQ(&smem[2*TPB*SLOT + tid*SLOT], Q[2]);
    WAIT_ASYNC(5);  buildQ(&smem[3*TPB*SLOT + tid*SLOT], Q[3]);

    asyncCopy72(slotB + 5u*bufStride, U + 1*(size_t)NS*18, gOff);    // U1 -> pong

    float b16[16];
#pragma unroll
    for (int k = 0; k < 16; k++) b16[k] = beta[k];

    // mu = 0 (U0 in ping)
    WAIT_ASYNC(5);
    doMu(Q, b16, 0, &smem[4*TPB*SLOT + tid*SLOT], out + (size_t)(0*NS + site)*18);
    WAIT_DS0();
    asyncCopy72(slotB + 4u*bufStride, U + 2*(size_t)NS*18, gOff);    // U2 -> ping

    // mu = 1 (U1 in pong)
    WAIT_ASYNC(5);
    doMu(Q, b16, 1, &smem[5*TPB*SLOT + tid*SLOT], out + (size_t)(1*NS + site)*18);
    WAIT_DS0();
    asyncCopy72(slotB + 5u*bufStride, U + 3*(size_t)NS*18, gOff);    // U3 -> pong

    // mu = 2 (U2 in ping)
    WAIT_ASYNC(5);
    doMu(Q, b16, 2, &smem[4*TPB*SLOT + tid*SLOT], out + (size_t)(2*NS + site)*18);

    // mu = 3 (U3 in pong)
    WAIT_ASYNC(0);
    doMu(Q, b16, 3, &smem[5*TPB*SLOT + tid*SLOT], out + (size_t)(3*NS + site)*18);
}

extern "C" void kernel_launch(void* const* d_in, const int* in_sizes, int n_in,
                              void* d_out, int out_size, void* d_ws, size_t ws_size,
                              hipStream_t stream) {
    (void)in_sizes; (void)n_in; (void)out_size; (void)d_ws; (void)ws_size;
    const float* U    = (const float*)d_in[0];   // complex64 as interleaved float
    const float* W    = (const float*)d_in[1];   // complex64 as interleaved float
    const float* beta = (const float*)d_in[2];   // float32 [4,4]
    float* out = (float*)d_out;                  // complex64 as interleaved float
    lge_exp_kernel<<<dim3(NS / TPB), dim3(TPB), 0, stream>>>(U, W, beta, out);
}